// HH_Gap_9045201126063
// MI455X (gfx1250) — compile-verified
//
#include <hip/hip_runtime.h>
#include <hip/hip_bf16.h>
#include <stdint.h>

// HH constants
#define HH_B 16
#define HH_N 4096
#define HH_L 512
#define T_CHUNK 64           // timesteps per async chunk
#define NBUF 3               // LDS buffers (triple buffering)
#define NCHUNK (HH_N / T_CHUNK)  // 64 chunks covering all 4096 z rows

__device__ __forceinline__ float frcp(float x) {
    // v_rcp_f32 + one Newton-Raphson step (~0.5 ulp) — far cheaper than IEEE div
    float r = __builtin_amdgcn_rcpf(x);
    return r * __builtin_fmaf(-x, r, 2.0f);
}

__device__ __forceinline__ float fexp2(float x) {
    return __builtin_amdgcn_exp2f(x);  // v_exp_f32
}

// sigmoid((V - VT)/KP) with VT=-20, KP=3
__device__ __forceinline__ float out_sigmoid(float V) {
    const float L2E = 1.44269504088896340736f;
    const float c = L2E / 3.0f;
    return frcp(1.0f + fexp2((-20.0f - V) * c));
}

// One Crank–Nicolson HH step. Updates V,m,n,h; returns new V.
__device__ __forceinline__ float hh_step(float z, float& V, float& m, float& n, float& h) {
    const float DT = 0.05f, DT2 = 0.025f;
    const float L2E = 1.44269504088896340736f;
    const float k9  = L2E / 9.0f;
    const float k12 = L2E / 12.0f;
    const float k197 = L2E / 19.7f;

    float m2 = m * m;
    float pow1 = 40.0f * m2 * m * h;          // gNa * m^3 * h
    float n2 = n * n;
    float pow2 = 35.0f * n2 * n2;             // gK * n^4
    float G = DT2 * (pow1 + pow2 + 0.3f);
    float E = __builtin_fmaf(pow1, 55.0f, __builtin_fmaf(pow2, -77.0f, 0.3f * -65.0f));
    float Vn = (__builtin_fmaf(V, 1.0f - G, DT * (E + z))) * frcp(1.0f + G);

    float vp35 = Vn + 35.0f;
    float e25  = fexp2((25.0f - Vn) * k9);    // exp((-V+25)/9)
    float em35 = fexp2(-vp35 * k9);           // exp((-V-35)/9)
    float ep35 = fexp2( vp35 * k9);           // exp((V+35)/9)
    float aN = 0.02f  * (Vn - 25.0f) * frcp(1.0f - e25);
    float aM = 0.182f * vp35         * frcp(1.0f - em35);
    float bM = -0.124f * vp35        * frcp(1.0f - ep35);
    float aH = 0.25f  * fexp2((-90.0f - Vn) * k12);
    float bN = 0.125f * fexp2((70.0f - Vn) * k197);
    float bH = 0.25f  * fexp2((34.0f + Vn) * k12);

    float sM = DT2 * (aM + bM);
    m = __builtin_fmaf(aM, DT, (1.0f - sM) * m) * frcp(1.0f + sM);
    float sN = DT2 * (aN + bN);
    n = __builtin_fmaf(aN, DT, (1.0f - sN) * n) * frcp(1.0f + sN);
    float sH = DT2 * (aH + bH);
    h = __builtin_fmaf(aH, DT, (1.0f - sH) * h) * frcp(1.0f + sH);

    V = Vn;
    return Vn;
}

// Issue one 64-row chunk: 16 x global_load_async_to_lds_b128.
// Lane i: row = i>>3, float column group = (i&7)*4 (16 B per lane, 4 rows/instr).
__device__ __forceinline__ void issue_chunk(const char* gbase, uint32_t lds) {
#pragma unroll
    for (int j = 0; j < 16; ++j) {
        uint64_t ga = (uint64_t)(uintptr_t)(gbase + (size_t)j * (4 * HH_L * 4)); // 4 rows = 8192 B
        uint32_t la = lds + (uint32_t)(j * 4 * 32 * 4);                          // 4 rows = 512 B in LDS
        asm volatile("global_load_async_to_lds_b128 %0, %1, off"
                     :: "v"(la), "v"(ga)
                     : "memory");
    }
}

__global__ __launch_bounds__(32)
void hh_scan_kernel(const float* __restrict__ z, float* __restrict__ out) {
    __shared__ float zbuf[NBUF * T_CHUNK * 32];   // 24 KB

    const int lane = threadIdx.x;                 // 0..31
    const int blk  = blockIdx.x;                  // 0..255
    const int b    = blk >> 4;                    // 16 blocks per batch entry
    const int l0   = (blk & 15) << 5;             // wave's 32-wide l window
    const int l    = l0 + lane;

    // loader mapping for b128 async copies
    const int rw = lane >> 3;                     // row within 4-row group
    const int cg = (lane & 7) << 2;               // float column offset (0,4,...,28)
    const char* gbase = (const char*)z +
        ((((size_t)b * HH_N + rw) * HH_L) + (size_t)l0 + (size_t)cg) * 4;

    uint32_t ldsRoot = (uint32_t)(uintptr_t)(&zbuf[0]);
    uint32_t myLds = ldsRoot + (uint32_t)((rw * 32 + cg) * 4);

    // state
    float V = -70.0f, m = 0.0f, n = 0.0f, h = 1.0f;

    float* op = out + ((size_t)b * HH_N) * HH_L + (size_t)l;
    __builtin_nontemporal_store(out_sigmoid(V), op);   // t = 0 row
    op += HH_L;

    // prime the pipeline: chunks 0..NBUF-1 in flight (48 async instrs <= 63 cap)
#pragma unroll
    for (int k = 0; k < NBUF; ++k)
        issue_chunk(gbase + (size_t)k * (T_CHUNK * HH_L * 4),
                    myLds + (uint32_t)(k * T_CHUNK * 32 * 4));

    for (int k = 0; k < NCHUNK; ++k) {
        // Guarantee chunk k landed: allow only the chunks issued *beyond* k to
        // remain outstanding. Steady state: 2 chunks (32 instrs) in flight.
        // Tail: the issue stream has ended, so the threshold must tighten or the
        // wait would pass while chunk k is still in flight.
        if (k <= NCHUNK - 3) {
            asm volatile("s_wait_asynccnt 32" ::: "memory");
        } else if (k == NCHUNK - 2) {
            asm volatile("s_wait_asynccnt 16" ::: "memory");
        } else { // k == NCHUNK - 1
            asm volatile("s_wait_asynccnt 0" ::: "memory");
        }

        const float* buf = &zbuf[(k % NBUF) * (T_CHUNK * 32)];
        const int nr = (k < NCHUNK - 1) ? T_CHUNK : (T_CHUNK - 1);  // 4095 z rows consumed

#pragma unroll 4
        for (int r = 0; r < nr; ++r) {
            float zv = buf[(r << 5) + lane];          // ds_load_2addr_b32 pairs
            float Vn = hh_step(zv, V, m, n, h);
            __builtin_nontemporal_store(out_sigmoid(Vn), op);
            op += HH_L;
        }

        if (k + NBUF < NCHUNK) {
            // close LDS WAR hazard: all DS reads of this slot retired before overwrite
            asm volatile("s_wait_dscnt 0" ::: "memory");
            issue_chunk(gbase + (size_t)(k + NBUF) * (T_CHUNK * HH_L * 4),
                        myLds + (uint32_t)(((k + NBUF) % NBUF) * T_CHUNK * 32 * 4));
        }
    }
}

extern "C" void kernel_launch(void* const* d_in, const int* in_sizes, int n_in,
                              void* d_out, int out_size, void* d_ws, size_t ws_size,
                              hipStream_t stream) {
    (void)in_sizes; (void)n_in; (void)d_ws; (void)ws_size; (void)out_size;
    const float* z = (const float*)d_in[0];
    float* out = (float*)d_out;
    // 8192 chains -> 256 single-wave (32-thread) workgroups spread across WGPs
    hh_scan_kernel<<<dim3(256), dim3(32), 0, stream>>>(z, out);
}